// HexPlane_8340826489618
// MI455X (gfx1250) — compile-verified
//
#include <hip/hip_runtime.h>
#include <stdint.h>

// ---------------------------------------------------------------------------
// HexPlane sampling for MI455X (gfx1250).
// Output: (N, 192) f32.  out[n, 8*k], k = plane*4+level (spatial k<12, temporal k>=12).
// Strategy: interleave tables to (H,W,C=8) in d_ws (b128 gathers), stage the
// two low-res spatial levels (120 KB) in LDS via async-to-LDS, stream output
// with non-temporal b128 stores.
// ---------------------------------------------------------------------------

typedef float v4f __attribute__((ext_vector_type(4)));
typedef int hp_v4i __attribute__((vector_size(16)));

#if defined(__has_builtin)
#  if __has_builtin(__builtin_amdgcn_global_load_async_to_lds_b128)
#    define HP_ASYNC_BUILTIN 1
#  endif
#  if __has_builtin(__builtin_amdgcn_s_wait_asynccnt)
#    define HP_WAITASYNC_BUILTIN 1
#  endif
#endif
#ifndef HP_ASYNC_BUILTIN
#define HP_ASYNC_BUILTIN 0
#endif
#ifndef HP_WAITASYNC_BUILTIN
#define HP_WAITASYNC_BUILTIN 0
#endif

__device__ __forceinline__ void hp_async_copy_b128(const float* gsrc, uint32_t lds_byte) {
#if HP_ASYNC_BUILTIN
  __builtin_amdgcn_global_load_async_to_lds_b128(
      (__attribute__((address_space(1))) hp_v4i*)(uintptr_t)gsrc,
      (__attribute__((address_space(3))) hp_v4i*)lds_byte,
      0, 0);
#else
  unsigned long long ga = (unsigned long long)(uintptr_t)gsrc;
  asm volatile("global_load_async_to_lds_b128 %0, %1, off"
               :: "v"(lds_byte), "v"(ga) : "memory");
#endif
}

__device__ __forceinline__ void hp_wait_async0() {
#if HP_WAITASYNC_BUILTIN
  __builtin_amdgcn_s_wait_asynccnt(0);
#else
  asm volatile("s_wait_asynccnt 0" ::: "memory");
#endif
}

// (3,C,H,W) channel-major -> (3,H,W,C) interleaved, C=8.
__global__ void hp_interleave(const float* __restrict__ src, float* __restrict__ dst,
                              int H, int W) {
  int i = blockIdx.x * blockDim.x + threadIdx.x;
  int total = 3 * H * W * 8;
  if (i >= total) return;
  int c = i & 7;
  int rest = i >> 3;
  int x = rest % W; rest /= W;
  int y = rest % H;
  int p = rest / H;
  dst[i] = src[((p * 8 + c) * H + y) * W + x];
}

__device__ __forceinline__ void hp_bilerp8(const float* bp,
                                           int i00, int i01, int i10, int i11,
                                           float w00, float w01, float w10, float w11,
                                           v4f& r0, v4f& r1) {
  v4f a0 = *(const v4f*)(bp + i00), a1 = *(const v4f*)(bp + i00 + 4);
  v4f b0 = *(const v4f*)(bp + i01), b1 = *(const v4f*)(bp + i01 + 4);
  v4f c0 = *(const v4f*)(bp + i10), c1 = *(const v4f*)(bp + i10 + 4);
  v4f d0 = *(const v4f*)(bp + i11), d1 = *(const v4f*)(bp + i11 + 4);
  r0 = w00 * a0 + w01 * b0 + w10 * c0 + w11 * d0;
  r1 = w00 * a1 + w01 * b1 + w10 * c1 + w11 * d1;
}

// Staged region = interleaved spatial levels 0 (16x16) and 1 (32x32), 3 planes:
// (2048 + 8192) * 3 = 30720 floats = 120 KB (CDNA5 allows up to 320 KB/WG).
#define HP_STAGE_FLOATS 30720

__global__ __launch_bounds__(256) void hexplane_sample(
    const float* __restrict__ xyz, const float* __restrict__ tarr,
    const float* __restrict__ bnd,
    const float* __restrict__ sp0, const float* __restrict__ sp1,
    const float* __restrict__ sp2, const float* __restrict__ sp3,
    const float* __restrict__ tp0, const float* __restrict__ tp1,
    const float* __restrict__ tp2, const float* __restrict__ tp3,
    const float* __restrict__ ws, float* __restrict__ out,
    int N, int interleaved)
{
  extern __shared__ float s_tab[];  // 120 KB when interleaved path is active

  if (interleaved) {
    const uint32_t lbase = (uint32_t)(uintptr_t)&s_tab[0];
    for (int i = threadIdx.x; i < HP_STAGE_FLOATS / 4; i += 256) {
      hp_async_copy_b128(ws + (size_t)i * 4, lbase + (uint32_t)i * 16u);
    }
    hp_wait_async0();
  }
  __syncthreads();

  const float lox = bnd[0], loy = bnd[1], loz = bnd[2];
  const float isx = 1.0f / (bnd[3] - lox);
  const float isy = 1.0f / (bnd[4] - loy);
  const float isz = 1.0f / (bnd[5] - loz);

  const int total = N * 24;
  for (int g = blockIdx.x * blockDim.x + threadIdx.x; g < total;
       g += gridDim.x * blockDim.x) {
    const int n = g / 24;
    const int k = g - n * 24;

    const float x = (xyz[3 * n + 0] - lox) * isx;
    const float y = (xyz[3 * n + 1] - loy) * isy;
    const float z = (xyz[3 * n + 2] - loz) * isz;

    float px, py;
    int Wd, Hd;
    const float* base = nullptr;
    bool useLds = false;
    int loff = 0;

    if (k < 12) {                       // spatial planes: align_corners = False
      const int p = k >> 2, l = k & 3;
      const int r = 16 << l;
      const float gx = (p == 2) ? y : x;   // {x, x, y}
      const float gy = (p == 0) ? y : z;   // {y, z, z}
      px = gx * (float)r - 0.5f;
      py = gy * (float)r - 0.5f;
      Wd = r; Hd = r;
      if (interleaved) {
        const int spoff = (l == 0) ? 0 : (l == 1) ? 6144 : (l == 2) ? 30720 : 129024;
        const int sppl  = 8 * r * r;
        if (l < 2) { useLds = true; loff = spoff + p * sppl; }
        else       { base = ws + spoff + p * sppl; }
      } else {
        const float* s = (l == 0) ? sp0 : (l == 1) ? sp1 : (l == 2) ? sp2 : sp3;
        base = s + (size_t)p * 8 * r * r;  // channel-major fallback
      }
    } else {                            // temporal planes: align_corners = True
      const int kk = k - 12, p = kk >> 2, l = kk & 3;
      const int r = 16 << l;
      const float tt = tarr[n];
      const float gy = (p == 0) ? x : (p == 1) ? y : z;  // {x, y, z}
      px = tt * 99.0f;                  // W = T = 100
      py = gy * (float)(r - 1);
      Wd = 100; Hd = r;
      if (interleaved) {
        const int tpoff = (l == 0) ? 522240 : (l == 1) ? 560640 : (l == 2) ? 637440 : 791040;
        const int tppl  = 800 * r;
        base = ws + tpoff + p * tppl;
      } else {
        const float* s = (l == 0) ? tp0 : (l == 1) ? tp1 : (l == 2) ? tp2 : tp3;
        base = s + (size_t)p * 8 * r * 100;
      }
    }

    px = fminf(fmaxf(px, 0.0f), (float)(Wd - 1));
    py = fminf(fmaxf(py, 0.0f), (float)(Hd - 1));
    const float fx = floorf(px), fy = floorf(py);
    const int x0 = (int)fx, y0 = (int)fy;
    const float wx = px - fx, wy = py - fy;
    const int x1 = min(x0 + 1, Wd - 1);
    const int y1 = min(y0 + 1, Hd - 1);
    const float w00 = (1.0f - wx) * (1.0f - wy);
    const float w01 = wx * (1.0f - wy);
    const float w10 = (1.0f - wx) * wy;
    const float w11 = wx * wy;

    v4f r0, r1;
    if (interleaved) {
      const int i00 = (y0 * Wd + x0) * 8, i01 = (y0 * Wd + x1) * 8;
      const int i10 = (y1 * Wd + x0) * 8, i11 = (y1 * Wd + x1) * 8;
      if (useLds) {
        hp_bilerp8(&s_tab[loff], i00, i01, i10, i11, w00, w01, w10, w11, r0, r1);
      } else {
        hp_bilerp8(base, i00, i01, i10, i11, w00, w01, w10, w11, r0, r1);
      }
    } else {
      const int HW = Hd * Wd;
      const int j00 = y0 * Wd + x0, j01 = y0 * Wd + x1;
      const int j10 = y1 * Wd + x0, j11 = y1 * Wd + x1;
      float acc[8];
#pragma unroll
      for (int c = 0; c < 8; ++c) {
        const float* pc = base + c * HW;
        acc[c] = w00 * pc[j00] + w01 * pc[j01] + w10 * pc[j10] + w11 * pc[j11];
      }
      r0 = (v4f){acc[0], acc[1], acc[2], acc[3]};
      r1 = (v4f){acc[4], acc[5], acc[6], acc[7]};
    }

    // 768 MB streaming output: NT stores keep tables resident in L2.
    float* op = out + (size_t)n * 192 + (size_t)k * 8;
    __builtin_nontemporal_store(r0, (v4f*)op);
    __builtin_nontemporal_store(r1, (v4f*)(op + 4));
  }
}

extern "C" void kernel_launch(void* const* d_in, const int* in_sizes, int n_in,
                              void* d_out, int out_size, void* d_ws, size_t ws_size,
                              hipStream_t stream) {
  (void)n_in; (void)out_size;
  const float* xyz = (const float*)d_in[0];
  const float* t   = (const float*)d_in[1];
  const float* bnd = (const float*)d_in[2];
  const float* sp[4] = {(const float*)d_in[3], (const float*)d_in[4],
                        (const float*)d_in[5], (const float*)d_in[6]};
  const float* tp[4] = {(const float*)d_in[7], (const float*)d_in[8],
                        (const float*)d_in[9], (const float*)d_in[10]};
  const int N = in_sizes[0] / 3;

  float* ws = (float*)d_ws;
  const size_t WS_NEED_BYTES = 1098240ull * 4ull;  // interleaved tables: ~4.4 MB
  const int interleaved = (d_ws != nullptr && ws_size >= WS_NEED_BYTES) ? 1 : 0;

  static const int    Rr[4]  = {16, 32, 64, 128};
  static const size_t spo[4] = {0, 6144, 30720, 129024};
  static const size_t tpo[4] = {522240, 560640, 637440, 791040};

  if (interleaved) {
    for (int l = 0; l < 4; ++l) {
      const int tot = 3 * Rr[l] * Rr[l] * 8;
      hp_interleave<<<(tot + 255) / 256, 256, 0, stream>>>(sp[l], ws + spo[l], Rr[l], Rr[l]);
    }
    for (int l = 0; l < 4; ++l) {
      const int tot = 3 * Rr[l] * 100 * 8;
      hp_interleave<<<(tot + 255) / 256, 256, 0, stream>>>(tp[l], ws + tpo[l], Rr[l], 100);
    }
  }

  const size_t shmem = interleaved ? (size_t)HP_STAGE_FLOATS * sizeof(float) : 0;
  hexplane_sample<<<2048, 256, shmem, stream>>>(
      xyz, t, bnd, sp[0], sp[1], sp[2], sp[3], tp[0], tp[1], tp[2], tp[3],
      ws, (float*)d_out, N, interleaved);
}